// DelayAndSum_10239202034037
// MI455X (gfx1250) — compile-verified
//
#include <hip/hip_runtime.h>
#include <stdint.h>

// DelayAndSum beamforming, MI455X (gfx1250, wave32).
//
// out[b,t] = (1/128) * sum_s x[b, t + d_s, s],  d_s = round(s*sin(30deg)/2)
// sin(pi/6) in double = 0.49999999999999994 -> no .5 ties -> d_s = (s+1)>>2.
//
// Memory-bound (819 MB read @ 23.3 TB/s ~ 35us). Strategy:
//  - Tensor Data Mover stages a [160 x 128] f32 tile into LDS per workgroup:
//      * D# pad fields insert 4 pad DWORDs per 128-DWORD row -> 132-float LDS
//        stride (16B-aligned rows, bank-conflict-free column gather)
//      * tensor_dim1 = T - t0 -> hardware zero-fill past the end of the signal
//  - sensor reduction via V_WMMA_F32_16X16X4_F32 with B = all 1/128
//    (row-sum as GEMM; accumulate K=128 in 32 WMMA steps per wave)

typedef float v2f  __attribute__((ext_vector_type(2)));
typedef float v8f  __attribute__((ext_vector_type(8)));
typedef unsigned int v4u __attribute__((ext_vector_type(4)));
typedef int   v4i  __attribute__((ext_vector_type(4)));
typedef int   v8i  __attribute__((ext_vector_type(8)));

#define NSENS      128
#define TILE_T     128                 // t-values per workgroup (8 waves x 16)
#define HALO       32                  // max delay d_127 = 32
#define ROWS       (TILE_T + HALO)     // 160 staged rows
#define LDS_STRIDE 132                 // 128 + 4 pad floats (528B rows)

#ifndef USE_TDM
#define USE_TDM 1                      // set 0 to fall back to float4 staging
#endif

__global__ __launch_bounds__(256)
void delay_sum_wmma_kernel(const float* __restrict__ x,
                           float* __restrict__ out,
                           int T)
{
    // Only static LDS object in this kernel -> allocated at LDS offset 0.
    __shared__ __align__(16) float lds[ROWS * LDS_STRIDE];

    const int b   = blockIdx.y;
    const int t0  = blockIdx.x * TILE_T;
    const int tid = threadIdx.x;

    const int lane  = tid & 31;
    const int wv    = tid >> 5;

    const float* src = x + (size_t)b * (size_t)T * NSENS;

#if USE_TDM
    // ------------- stage tile into LDS via Tensor Data Mover ----------------
    if (wv == 0) {
        const uint64_t gaddr = (uint64_t)(uintptr_t)(src + (size_t)t0 * NSENS);

        // D# group 0: flags | lds_addr | global_addr | type
        v4u g0;
        g0[0] = 1u;                                   // count=1 (valid), no gather
        g0[1] = 0u;                                   // lds_addr = 0 (array at base)
        g0[2] = (uint32_t)(gaddr & 0xFFFFFFFFu);      // global_addr[31:0]
        g0[3] = (uint32_t)((gaddr >> 32) & 0x1FFFFFFu)// global_addr[56:32]
              | (2u << 30);                           // type = 2 ("image")

        // D# group 1: sizes / padding / strides
        const uint32_t td0 = NSENS;                   // tensor dim0 (elements)
        const uint32_t td1 = (uint32_t)(T - t0);      // rows left -> HW zero-fill OOB
        const uint32_t tl0 = NSENS;                   // tile dim0
        const uint32_t tl1 = ROWS;                    // tile dim1
        const uint64_t st0 = NSENS;                   // dim0 stride (elements)
        v8i g1;
        g1[0] = (int)((2u << 16)                      // data_size = 4 bytes
                    | (1u << 20)                      // pad_enable
                    | (6u << 22)                      // pad_interval: every 128 DW
                    | (3u << 25));                    // pad_amount: 4 DWORDs
        g1[1] = (int)((td0 & 0xFFFFu) << 16);         // abar_addr=0 | td0[15:0]
        g1[2] = (int)((td0 >> 16) | ((td1 & 0xFFFFu) << 16));
        g1[3] = (int)((td1 >> 16) | (tl0 << 16));
        g1[4] = (int)(tl1);                           // tile_dim1 | tile_dim2=0
        g1[5] = (int)(uint32_t)(st0 & 0xFFFFFFFFu);
        g1[6] = (int)(uint32_t)(st0 >> 32);           // | dim1_stride lo16 = 0
        g1[7] = 0;                                    // dim1_stride hi = 0

        const v4i gz = {0, 0, 0, 0};                  // groups 2/3: dims unused
#if defined(__clang_major__) && (__clang_major__ >= 23)
        const v8i gz8 = {0, 0, 0, 0, 0, 0, 0, 0};
        __builtin_amdgcn_tensor_load_to_lds(g0, g1, gz, gz, gz8, 0);
#else
        __builtin_amdgcn_tensor_load_to_lds(g0, g1, gz, gz, 0);
#endif
        __builtin_amdgcn_s_wait_tensorcnt(0);
    }
    __syncthreads();
#else
    // ------------- fallback: coalesced float4 staging -----------------------
    for (int i = tid; i < ROWS * (NSENS / 4); i += 256) {
        const int r  = i >> 5;
        const int c4 = i & 31;
        const int tg = t0 + r;
        float4 v = make_float4(0.f, 0.f, 0.f, 0.f);
        if (tg < T) {
            v = *(const float4*)(src + (size_t)tg * NSENS + (c4 << 2));
        }
        *(float4*)(&lds[r * LDS_STRIDE + (c4 << 2)]) = v;
    }
    __syncthreads();
#endif

    // The TDM writes `lds` behind the compiler's back; without an escape +
    // memory clobber, LLVM proves the array is never stored and folds the
    // LDS loads below to undef (observed: ds=0). Zero instructions emitted.
    {
        float* lds_escape = &lds[0];
        asm volatile("" : "+v"(lds_escape) : : "memory");
    }

    // ---------------- per-wave 16x(K=128) row-sum via WMMA -------------------
    const int m     = lane & 15;        // A-matrix row (M)
    const int khalf = lane >> 4;        // 0 -> K 0,1 ; 1 -> K 2,3
    const int tloc  = wv * 16;          // this wave's t offset within tile

    v8f c = {};                                        // f32 accumulator (D/C)
    const v2f bmat = { 1.0f / 128.0f, 1.0f / 128.0f }; // B = all 1/128

#pragma unroll
    for (int kb = 0; kb < 32; ++kb) {
        // sensors in this K-block: s = 4*kb + k, k = 0..3
        // delays: k in {0,1,2} -> kb ; k == 3 -> kb + 1
        const int s0 = (kb << 2) + (khalf << 1);
        const float a0 = lds[(tloc + m + kb)         * LDS_STRIDE + s0];
        const float a1 = lds[(tloc + m + kb + khalf) * LDS_STRIDE + s0 + 1];
        const v2f a = { a0, a1 };
        c = __builtin_amdgcn_wmma_f32_16x16x4_f32(
                /*neg_a=*/false, a,
                /*neg_b=*/false, bmat,
                /*c_mod=*/(short)0, c,
                /*reuse_a=*/false, /*reuse_b=*/false);
    }

    // ---------------- store: all D columns identical, take N == 0 ------------
    // D layout: VGPR r, lanes 0-15 -> M=r ; lanes 16-31 -> M=8+r
    if ((lane & 15) == 0) {
        const int tbase = t0 + tloc + (khalf << 3);
        float* dst = out + (size_t)b * T;
#pragma unroll
        for (int r = 0; r < 8; ++r) {
            const int t = tbase + r;
            if (t < T) dst[t] = c[r];
        }
    }
}

extern "C" void kernel_launch(void* const* d_in, const int* in_sizes, int n_in,
                              void* d_out, int out_size, void* d_ws, size_t ws_size,
                              hipStream_t stream)
{
    const float* x  = (const float*)d_in[0];
    float* out      = (float*)d_out;

    const int B = 16;
    const int T = in_sizes[0] / (B * NSENS);   // 100000

    dim3 grid((T + TILE_T - 1) / TILE_T, B);
    delay_sum_wmma_kernel<<<grid, 256, 0, stream>>>(x, out, T);
}